// LTCCell_57114475102570
// MI455X (gfx1250) — compile-verified
//
#include <hip/hip_runtime.h>
#include <cstdint>

// Problem constants (fixed by the reference)
#define LTC_B        8192
#define LTC_IN       64
#define LTC_U        128
#define LTC_UNFOLDS  3
#define LTC_EPS      1e-8f

// Tiling
#define THREADS   1024           // 32 waves (wave32) -> 8 waves / SIMD32
#define ROWS      32             // batch rows per workgroup
#define TEAMS     8              // THREADS / LTC_U
#define RPT       4              // rows per team (ROWS / TEAMS)

// ---- sigmoid: prefer gfx1250 v_tanh_f32 (1 TRANS) over exp+rcp (2 TRANS) ----
// sigmoid(x) = 0.5*tanh(x/2) + 0.5, with the 0.5 folded into the packed params.
#if __has_builtin(__builtin_amdgcn_tanhf)
  #define KSCALE 0.5f            // t = 0.5*x
  __device__ __forceinline__ float gate_from_t(float t) {
    return fmaf(__builtin_amdgcn_tanhf(t), 0.5f, 0.5f);
  }
#elif defined(__gfx1250__)
  #define KSCALE 0.5f            // t = 0.5*x ; guaranteed v_tanh_f32 via asm.
  __device__ __forceinline__ float gate_from_t(float t) {
    float r;
    // v_nop covers the ISA §7.4 trans-op hazard (compiler can't see into asm).
    asm volatile("v_tanh_f32 %0, %1\n\tv_nop" : "=v"(r) : "v"(t));
    return fmaf(r, 0.5f, 0.5f);
  }
#elif __has_builtin(__builtin_amdgcn_exp2f) && __has_builtin(__builtin_amdgcn_rcpf)
  #define KSCALE 1.4426950408889634f   // t = log2(e)*x ; sigmoid = rcp(1+2^-t)
  __device__ __forceinline__ float gate_from_t(float t) {
    return __builtin_amdgcn_rcpf(1.0f + __builtin_amdgcn_exp2f(-t));
  }
#else
  #define KSCALE 1.0f
  __device__ __forceinline__ float gate_from_t(float t) {
    return 1.0f / (1.0f + __expf(-t));
  }
#endif

#if __has_builtin(__builtin_amdgcn_rcpf)
  __device__ __forceinline__ float fast_rcp(float x) { return __builtin_amdgcn_rcpf(x); }
#else
  __device__ __forceinline__ float fast_rcp(float x) { return 1.0f / x; }
#endif

// ---- CDNA5 async global->LDS copy (ASYNCcnt path, CDNA5 ISA) ----
// vdst = LDS byte offset (flat addr low 32 bits == LDS offset per ISA 10.2),
// vaddr = 64-bit global address.
__device__ __forceinline__ void async_copy_b128(uint32_t lds_off, const void* gaddr) {
  asm volatile("global_load_async_to_lds_b128 %0, %1, off"
               :: "v"(lds_off), "v"(gaddr) : "memory");
}
__device__ __forceinline__ void wait_async0() {
  asm volatile("s_wait_asynccnt 0x0" ::: "memory");
}
__device__ __forceinline__ uint32_t lds_off_of(const void* p) {
  return (uint32_t)(uintptr_t)p;   // flat LDS addr[31:0] == workgroup LDS offset
}

__global__ __launch_bounds__(THREADS, 1)
void ltc_cell_kernel(const float* __restrict__ inputs,   // [B, IN]
                     const float* __restrict__ state,    // [B, U]
                     const float* __restrict__ s_mu,     // [IN, U]
                     const float* __restrict__ s_sigma,  // [IN, U]
                     const float* __restrict__ s_W,      // [IN, U]
                     const float* __restrict__ s_erev,   // [IN, U]
                     const float* __restrict__ mu,       // [U, U]
                     const float* __restrict__ sigma,    // [U, U]
                     const float* __restrict__ W,        // [U, U]
                     const float* __restrict__ erev,     // [U, U]
                     const float* __restrict__ vleak,    // [U]
                     const float* __restrict__ gleak,    // [U]
                     const float* __restrict__ cm_t,     // [U]
                     float* __restrict__ out)            // [B, U]
{
  // 256 KB packed params: (KSCALE*sigma, KSCALE*sigma*mu, W*erev, W) at [j*U+u].
  // Sensory phase aliases the front 128 KB before being overwritten.
  __shared__ float4 pk[LTC_U * LTC_U];        // 256 KB
  __shared__ float  vtile[ROWS * LTC_U];      //  16 KB
  __shared__ float  intile[ROWS * LTC_IN];    //   8 KB

  const int tid  = threadIdx.x;
  const int u    = tid & (LTC_U - 1);
  const int team = tid >> 7;                  // 0..7
  const int b0   = blockIdx.x * ROWS;

  // per-u leak scalars
  const float cm = cm_t[u];
  const float gl = gleak[u];
  const float gv = gl * vleak[u];

  // ---- Phase 0: async-stage state & input tiles into LDS (ASYNCcnt) ----
  {
    const float4* gst = (const float4*)(state + (size_t)b0 * LTC_U);
    async_copy_b128(lds_off_of(vtile) + (uint32_t)tid * 16u, gst + tid);   // 1024 x 16B = 16KB
    if (tid < (ROWS * LTC_IN / 4)) {                                       // 512 x 16B = 8KB
      const float4* gin = (const float4*)(inputs + (size_t)b0 * LTC_IN);
      async_copy_b128(lds_off_of(intile) + (uint32_t)tid * 16u, gin + tid);
    }
  }

  // ---- Phase 1: pack sensory params into LDS (coalesced global reads) ----
  #pragma unroll
  for (int k = 0; k < (LTC_IN * LTC_U) / THREADS; ++k) {   // 8 iters
    const int e = k * THREADS + tid;
    const float sg = s_sigma[e];
    const float m  = s_mu[e];
    const float w  = s_W[e];
    const float er = s_erev[e];
    pk[e] = make_float4(KSCALE * sg, KSCALE * sg * m, w * er, w);
  }
  wait_async0();
  __syncthreads();

  // ---- Phase 2: sensory conductance sums (per team-row, per u) ----
  float ns[RPT], nd[RPT];
  #pragma unroll
  for (int r = 0; r < RPT; ++r) {
    const int row = team * RPT + r;
    float accn = 0.0f, accd = 0.0f;
    const float4* vin4 = (const float4*)(&intile[row * LTC_IN]);
    #pragma unroll 2
    for (int i4 = 0; i4 < LTC_IN / 4; ++i4) {
      const float4 vq = vin4[i4];                    // broadcast within team
      const float vv[4] = {vq.x, vq.y, vq.z, vq.w};
      #pragma unroll
      for (int s = 0; s < 4; ++s) {
        const float4 p = pk[(i4 * 4 + s) * LTC_U + u];
        const float t = fmaf(p.x, vv[s], -p.y);      // K*sigma*(v - mu)
        const float g = gate_from_t(t);
        accn = fmaf(p.z, g, accn);                   // + W*erev*g
        accd = fmaf(p.w, g, accd);                   // + W*g
      }
    }
    ns[r] = accn; nd[r] = accd;
  }
  __syncthreads();

  // ---- Phase 3: pack recurrent params (overwrites sensory region) ----
  #pragma unroll
  for (int k = 0; k < (LTC_U * LTC_U) / THREADS; ++k) {    // 16 iters
    const int e = k * THREADS + tid;
    const float sg = sigma[e];
    const float m  = mu[e];
    const float w  = W[e];
    const float er = erev[e];
    pk[e] = make_float4(KSCALE * sg, KSCALE * sg * m, w * er, w);
  }
  __syncthreads();

  // ---- Phase 4: ODE unfolds ----
  float vcur[RPT];
  #pragma unroll
  for (int r = 0; r < RPT; ++r) vcur[r] = vtile[(team * RPT + r) * LTC_U + u];

  for (int it = 0; it < LTC_UNFOLDS; ++it) {
    float vnew[RPT];
    #pragma unroll
    for (int r = 0; r < RPT; ++r) {
      const int row = team * RPT + r;
      float accn = ns[r], accd = nd[r];
      const float4* vrow4 = (const float4*)(&vtile[row * LTC_U]);
      #pragma unroll 2
      for (int j4 = 0; j4 < LTC_U / 4; ++j4) {       // 32 iters
        const float4 vq = vrow4[j4];                 // team-broadcast LDS read
        const float vv[4] = {vq.x, vq.y, vq.z, vq.w};
        #pragma unroll
        for (int s = 0; s < 4; ++s) {
          const float4 p = pk[(j4 * 4 + s) * LTC_U + u];
          const float t = fmaf(p.x, vv[s], -p.y);
          const float g = gate_from_t(t);
          accn = fmaf(p.z, g, accn);
          accd = fmaf(p.w, g, accd);
        }
      }
      const float num = fmaf(cm, vcur[r], gv) + accn;
      const float den = cm + gl + accd + LTC_EPS;
      vnew[r] = num * fast_rcp(den);
    }
    __syncthreads();                                 // all team reads done
    #pragma unroll
    for (int r = 0; r < RPT; ++r) {
      vtile[(team * RPT + r) * LTC_U + u] = vnew[r];
      vcur[r] = vnew[r];
    }
    __syncthreads();                                 // writes visible
  }

  // ---- Phase 5: write result ----
  #pragma unroll
  for (int r = 0; r < RPT; ++r) {
    const int b = b0 + team * RPT + r;
    out[(size_t)b * LTC_U + u] = vcur[r];
  }
}

extern "C" void kernel_launch(void* const* d_in, const int* in_sizes, int n_in,
                              void* d_out, int out_size, void* d_ws, size_t ws_size,
                              hipStream_t stream) {
  (void)in_sizes; (void)n_in; (void)out_size; (void)d_ws; (void)ws_size;
  const float* inputs  = (const float*)d_in[0];
  const float* state   = (const float*)d_in[1];
  const float* s_mu    = (const float*)d_in[2];
  const float* s_sigma = (const float*)d_in[3];
  const float* s_W     = (const float*)d_in[4];
  const float* s_erev  = (const float*)d_in[5];
  const float* mu      = (const float*)d_in[6];
  const float* sigma   = (const float*)d_in[7];
  const float* W       = (const float*)d_in[8];
  const float* erev    = (const float*)d_in[9];
  const float* vleak   = (const float*)d_in[10];
  const float* gleak   = (const float*)d_in[11];
  const float* cm_t    = (const float*)d_in[12];
  float* out = (float*)d_out;

  dim3 grid(LTC_B / ROWS);   // 256 workgroups, one WGP-sized tile each
  ltc_cell_kernel<<<grid, THREADS, 0, stream>>>(
      inputs, state, s_mu, s_sigma, s_W, s_erev,
      mu, sigma, W, erev, vleak, gleak, cm_t, out);
}